// HeavyEncoderLayer_74388833566995
// MI455X (gfx1250) — compile-verified
//
#include <hip/hip_runtime.h>
#include <hip/hip_bf16.h>
#include <math.h>

#define N_NODES 8000
#define N_EDGES 24000

typedef __attribute__((ext_vector_type(2))) float v2f;
typedef __attribute__((ext_vector_type(8))) float v8f;

// ---- constants (match reference scalings) ----
constexpr float INV3     = 0.57735026918962576f;   // 1/sqrt(3)
constexpr float PW_MSG0  = 0.10206207261596575f;   // sqrt(1/96)
constexpr float PW_MSG1  = 0.17677669529663688f;   // sqrt(3/96)
constexpr float PW_G01   = 0.022097086912079608f;  // sqrt(1/2048)
constexpr float PW_G2    = 0.03423265984407288f;   // sqrt(3/2560)
constexpr float PW_H0    = 0.031008683647302115f;  // sqrt(1/1040)
constexpr float PW_H1    = 0.30618621784789724f;   // sqrt(3/32)
constexpr float RS_MLP1  = 0.5f;                   // 1/sqrt(4)
constexpr float RS_MLP2  = 0.08838834764831845f;   // 1/sqrt(128)
constexpr float RS_L1    = 0.17677669529663688f;   // 1/sqrt(32)
constexpr float CST_SILU = 1.6790425f;
constexpr float CST_SIG  = 1.8467355f;
constexpr float CST_TANH = 1.5939103f;

// fp32 WMMA 16x16x4: D = A(16x4) x B(4x16) + C, wave32.
__device__ __forceinline__ v8f wmma4(v2f a, v2f b, v8f c) {
  return __builtin_amdgcn_wmma_f32_16x16x4_f32(false, a, false, b, (short)0, c,
                                               false, false);
}

// ============================ K0: init ============================
__global__ __launch_bounds__(256) void init_kernel(float* nm, float* sums, float* cnt) {
  int i = blockIdx.x * 256 + threadIdx.x;
  if (i < N_NODES * 112) { nm[i] = 0.f; sums[i] = 0.f; }
  if (i < N_NODES) cnt[i] = 0.f;
}

// ===================== K1: fused edge kernel ======================
// 16 edges per workgroup, 8 waves. Fuses:
//   h = CST_SILU*silu(ea@W1/2);  w = h@W2  (WMMA f32, never hits HBM);
//   per-edge tensor-product contraction of w with x[src];
//   atomic scatter into nm[dst].
__global__ __launch_bounds__(256) void edge_kernel(
    const float* __restrict__ x, const float* __restrict__ ea,
    const float* __restrict__ W1, const float* __restrict__ W2,
    const int* __restrict__ eidx, float* __restrict__ nm) {
  __shared__ float sh_h[16][128];
  __shared__ float sh_x0[16][64];
  __shared__ float sh_x1[16][96];   // [u][m] -> u*3+m
  __shared__ float sh_q1[16][32];   // sum_m x1[u,m]*a1[m]
  __shared__ float sh_a[16][4];
  __shared__ int   sh_dst[16];
  // per-wave private accumulators: [A000(16) | A110(16) | V011(32) | V101(96)]
  __shared__ float bank[8][16][160];

  const int tid  = threadIdx.x;
  const int wave = tid >> 5;
  const int lane = tid & 31;
  const int half = lane >> 4;
  const int er   = lane & 15;
  const int e0   = blockIdx.x * 16;

  if (tid < 64) { int e = tid >> 2, d = tid & 3; sh_a[e][d] = ea[(e0 + e) * 4 + d]; }
  if (tid < 16) sh_dst[tid] = eidx[N_EDGES + e0 + tid];
  for (int idx = tid; idx < 16 * 160; idx += 256) {
    int e = idx / 160, j = idx - e * 160;
    int s = eidx[e0 + e];
    float v = x[(size_t)s * 160 + j];
    if (j < 64) sh_x0[e][j] = v; else sh_x1[e][j - 64] = v;
  }
  for (int idx = tid; idx < 8 * 16 * 160; idx += 256) ((float*)bank)[idx] = 0.f;
  __syncthreads();

  for (int idx = tid; idx < 16 * 32; idx += 256) {
    int e = idx >> 5, u = idx & 31;
    sh_q1[e][u] = sh_x1[e][u * 3 + 0] * sh_a[e][1] +
                  sh_x1[e][u * 3 + 1] * sh_a[e][2] +
                  sh_x1[e][u * 3 + 2] * sh_a[e][3];
  }
  for (int idx = tid; idx < 16 * 128; idx += 256) {
    int e = idx >> 7, r = idx & 127;
    float s = 0.f;
#pragma unroll
    for (int d = 0; d < 4; d++) s += sh_a[e][d] * W1[d * 128 + r];
    s *= RS_MLP1;
    float sig = 1.f / (1.f + __expf(-s));
    sh_h[e][r] = CST_SILU * s * sig;
  }
  __syncthreads();

  // Preload full A (16x128) for this wave into VGPRs (ISA 16x4 f32 layout).
  v2f A[32];
#pragma unroll
  for (int kk = 0; kk < 32; kk++) {
    A[kk].x = sh_h[er][kk * 4 + 2 * half + 0];
    A[kk].y = sh_h[er][kk * 4 + 2 * half + 1];
  }

  float* mybank = &bank[wave][0][0];
  for (int tile = wave; tile < 288; tile += 8) {
    int col = tile * 16 + er;
    if (tile + 8 < 288) __builtin_prefetch(W2 + (tile + 8) * 16 + er, 0, 1);
    v8f c = {};
    const float* Bp = W2 + (2 * half) * 4608 + col;
#pragma unroll
    for (int kk = 0; kk < 32; kk++) {
      v2f b;
      b.x = Bp[(kk * 4 + 0) * 4608];
      b.y = Bp[(kk * 4 + 1) * 4608];
      c = wmma4(A[kk], b, c);
    }
    // epilogue: contract w-tile with per-edge features into wave-private bank
    int cc = col;  // global column for this lane
#pragma unroll
    for (int i = 0; i < 8; i++) {
      int e = half ? (i + 8) : i;
      float v = c[i];
      if (cc < 1024) {                    // s000: [u][w16], coef xs0[u]
        int u = cc >> 4, w = cc & 15;
        mybank[e * 160 + w] += sh_x0[e][u] * v;
      } else if (cc < 3072) {             // s011: [u][w32], coef xs0[u]
        int t = cc - 1024; int u = t >> 5, w = t & 31;
        mybank[e * 160 + 32 + w] += sh_x0[e][u] * v;
      } else if (cc < 4096) {             // s101: [u][w32], coef xs1[u][m]
        int t = cc - 3072; int u = t >> 5, w = t & 31;
        float* p = &mybank[e * 160 + 64 + w * 3];
        p[0] += sh_x1[e][u * 3 + 0] * v;
        p[1] += sh_x1[e][u * 3 + 1] * v;
        p[2] += sh_x1[e][u * 3 + 2] * v;
      } else {                            // s110: [u][w16], coef q1[u]
        int t = cc - 4096; int u = t >> 4, w = t & 15;
        mybank[e * 160 + 16 + w] += sh_q1[e][u] * v;
      }
    }
  }
  __syncthreads();

  // reduce 8 wave-banks, finalize msg, atomic scatter to nm[dst]
  for (int idx = tid; idx < 16 * 112; idx += 256) {
    int e = idx / 112, j = idx - e * 112;
    float v;
    if (j < 16) {
      float sA = 0.f, sB = 0.f;
      for (int wv = 0; wv < 8; wv++) { sA += bank[wv][e][j]; sB += bank[wv][e][16 + j]; }
      v = PW_MSG0 * RS_MLP2 * (sh_a[e][0] * sA + INV3 * sB);
    } else {
      int j2 = j - 16; int w = j2 / 3, m = j2 - w * 3;
      float s011 = 0.f, s101 = 0.f;
      for (int wv = 0; wv < 8; wv++) { s011 += bank[wv][e][32 + w]; s101 += bank[wv][e][64 + j2]; }
      v = PW_MSG1 * INV3 * RS_MLP2 * (sh_a[e][1 + m] * s011 + sh_a[e][0] * s101);
    }
    unsafeAtomicAdd(&nm[(size_t)sh_dst[e] * 112 + j], v);
  }
}

// ===================== K2: fused node-gate kernel ======================
// 16 nodes per workgroup. WMMA GEMMs with on-the-fly outer-product A:
//  path I : [16,2048]x[2048,48]  -> gs(16) | gg(32)
//  path II: per m: [16,2560]x[2560,32] -> gv(:, :, m)
// K split across 8 waves, partial C tiles reduced in LDS.
__global__ __launch_bounds__(256) void node_gate_kernel(
    const float* __restrict__ x, const float* __restrict__ nm,
    const float* __restrict__ Wg000, const float* __restrict__ Wg110,
    const float* __restrict__ Wg001, const float* __restrict__ Wg111,
    const float* __restrict__ Wg012, const float* __restrict__ Wg102,
    const int* __restrict__ z, const int* __restrict__ canon,
    float* __restrict__ xg, float* __restrict__ sums, float* __restrict__ cnt) {
  __shared__ float sm0[16][16];
  __shared__ float sm1[16][96];
  __shared__ float sx0[16][64];
  __shared__ float sx1[16][96];
  __shared__ float part[8][9][256];  // [wave][task][i*32+lane]

  const int tid  = threadIdx.x;
  const int wave = tid >> 5;
  const int lane = tid & 31;
  const int half = lane >> 4;
  const int er   = lane & 15;
  const int n0   = blockIdx.x * 16;

  for (int idx = tid; idx < 16 * 112; idx += 256) {
    int e = idx / 112, j = idx - e * 112;
    float v = nm[(size_t)(n0 + e) * 112 + j];
    if (j < 16) sm0[e][j] = v; else sm1[e][j - 16] = v;
  }
  for (int idx = tid; idx < 16 * 160; idx += 256) {
    int e = idx / 160, j = idx - e * 160;
    float v = x[(size_t)(n0 + e) * 160 + j];
    if (j < 64) sx0[e][j] = v; else sx1[e][j - 64] = v;
  }
  __syncthreads();

  // ---- Path I: K=2048 (m0 (x) x0 | INV3 * (m1 . x1)) ----
  v8f cI0 = {}, cI1 = {}, cI2 = {};
  for (int k = wave * 4; k < 2048; k += 32) {
    int kk = k + 2 * half;  // even, never straddles 1024
    v2f a, b0, b1, b2;
    if (kk < 1024) {
      a.x = sm0[er][kk >> 6] * sx0[er][kk & 63];
      a.y = sm0[er][(kk + 1) >> 6] * sx0[er][(kk + 1) & 63];
      const float* p0 = Wg000 + kk * 16 + er;
      const float* p1 = Wg001 + kk * 32 + er;
      b0.x = p0[0];  b0.y = p0[16];
      b1.x = p1[0];  b1.y = p1[32];
      b2.x = p1[16]; b2.y = p1[48];
    } else {
      int j = kk - 1024;
      int u0 = j >> 5, v0 = j & 31, u1 = (j + 1) >> 5, v1 = (j + 1) & 31;
      a.x = INV3 * (sm1[er][u0*3+0]*sx1[er][v0*3+0] + sm1[er][u0*3+1]*sx1[er][v0*3+1] + sm1[er][u0*3+2]*sx1[er][v0*3+2]);
      a.y = INV3 * (sm1[er][u1*3+0]*sx1[er][v1*3+0] + sm1[er][u1*3+1]*sx1[er][v1*3+1] + sm1[er][u1*3+2]*sx1[er][v1*3+2]);
      const float* p0 = Wg110 + j * 16 + er;
      const float* p1 = Wg111 + j * 32 + er;
      b0.x = p0[0];  b0.y = p0[16];
      b1.x = p1[0];  b1.y = p1[32];
      b2.x = p1[16]; b2.y = p1[48];
    }
    cI0 = wmma4(a, b0, cI0);
    cI1 = wmma4(a, b1, cI1);
    cI2 = wmma4(a, b2, cI2);
  }
#pragma unroll
  for (int i = 0; i < 8; i++) {
    part[wave][0][i * 32 + lane] = cI0[i];
    part[wave][1][i * 32 + lane] = cI1[i];
    part[wave][2][i * 32 + lane] = cI2[i];
  }

  // ---- Path II: per m, K=2560 (m0 (x) x1_m | m1_m (x) x0) ----
  v8f cA[3], cB[3];
#pragma unroll
  for (int m = 0; m < 3; m++) { cA[m] = (v8f){}; cB[m] = (v8f){}; }
  for (int k = wave * 4; k < 2560; k += 32) {
    int kk = k + 2 * half;  // even, never straddles 512
#pragma unroll
    for (int m = 0; m < 3; m++) {
      v2f a, bl, bh;
      if (kk < 512) {
        a.x = sm0[er][kk >> 5] * sx1[er][(kk & 31) * 3 + m];
        a.y = sm0[er][(kk + 1) >> 5] * sx1[er][((kk + 1) & 31) * 3 + m];
        const float* p = Wg012 + kk * 32 + er;
        bl.x = p[0];  bl.y = p[32];
        bh.x = p[16]; bh.y = p[48];
      } else {
        int j = kk - 512;
        a.x = sm1[er][(j >> 6) * 3 + m] * sx0[er][j & 63];
        a.y = sm1[er][((j + 1) >> 6) * 3 + m] * sx0[er][(j + 1) & 63];
        const float* p = Wg102 + j * 32 + er;
        bl.x = p[0];  bl.y = p[32];
        bh.x = p[16]; bh.y = p[48];
      }
      cA[m] = wmma4(a, bl, cA[m]);
      cB[m] = wmma4(a, bh, cB[m]);
    }
  }
#pragma unroll
  for (int m = 0; m < 3; m++)
#pragma unroll
    for (int i = 0; i < 8; i++) {
      part[wave][3 + m * 2][i * 32 + lane] = cA[m][i];
      part[wave][4 + m * 2][i * 32 + lane] = cB[m][i];
    }
  __syncthreads();

  // reduce partials, gate, write xg, masked atomic segment-sum
  for (int idx = tid; idx < 16 * 112; idx += 256) {
    int e = idx / 112, j = idx - e * 112;
    int i = e & 7, lb = (e >= 8) ? 16 : 0;
    float outv;
    if (j < 16) {
      int pi = i * 32 + lb + j;
      float gs = 0.f;
      for (int wv = 0; wv < 8; wv++) gs += part[wv][0][pi];
      gs *= PW_G01;
      outv = CST_SIG / (1.f + __expf(-gs));
    } else {
      int j2 = j - 16; int w = j2 / 3, m = j2 - w * 3;
      int tg = (w < 16) ? 1 : 2;
      int tv = 3 + m * 2 + ((w >= 16) ? 1 : 0);
      int pi = i * 32 + lb + (w & 15);
      float gg = 0.f, gv = 0.f;
      for (int wv = 0; wv < 8; wv++) { gg += part[wv][tg][pi]; gv += part[wv][tv][pi]; }
      gg *= PW_G01;
      gv *= PW_G2 * INV3;
      outv = CST_TANH * tanhf(gg) * gv;
    }
    int n = n0 + e;
    xg[(size_t)n * 112 + j] = outv;
    if (z[n] > 1) unsafeAtomicAdd(&sums[(size_t)canon[n] * 112 + j], outv);
  }
  if (tid < 16) {
    int n = n0 + tid;
    if (z[n] > 1) unsafeAtomicAdd(&cnt[canon[n]], 1.0f);
  }
}

// ===================== K3: per-node th ======================
__global__ __launch_bounds__(256) void th_kernel(
    const float* __restrict__ sums, const float* __restrict__ cnt,
    const float* __restrict__ Wa, const float* __restrict__ Wb,
    const float* __restrict__ Wc, const float* __restrict__ Wd,
    float* __restrict__ th) {
  __shared__ float sh1[256][96];
  int t = threadIdx.x;
  int n = blockIdx.x * 256 + t;
  bool act = (n < N_NODES);
  float h0[16];
  if (act) {
    float ic = 1.f / fmaxf(cnt[n], 1.f);
#pragma unroll
    for (int j = 0; j < 16; j++) h0[j] = sums[(size_t)n * 112 + j] * ic;
    for (int j = 0; j < 96; j++) sh1[t][j] = sums[(size_t)n * 112 + 16 + j] * ic;
  } else {
#pragma unroll
    for (int j = 0; j < 16; j++) h0[j] = 0.f;
    for (int j = 0; j < 96; j++) sh1[t][j] = 0.f;
  }
  float td[16];
#pragma unroll
  for (int w = 0; w < 16; w++) td[w] = 0.f;
  for (int p = 0; p < 32; p++) {
    float hp0 = sh1[t][p * 3 + 0], hp1 = sh1[t][p * 3 + 1], hp2 = sh1[t][p * 3 + 2];
    for (int q = 0; q < 32; q++) {
      float tmp = hp0 * sh1[t][q * 3 + 0] + hp1 * sh1[t][q * 3 + 1] + hp2 * sh1[t][q * 3 + 2];
      const float* wd = &Wd[(p * 32 + q) * 16];
#pragma unroll
      for (int w = 0; w < 16; w++) td[w] = fmaf(wd[w], tmp, td[w]);
    }
  }
  if (act) {
#pragma unroll
    for (int u = 0; u < 16; u++) {
      float s = 0.f;
#pragma unroll
      for (int v = 0; v < 16; v++) s = fmaf(Wa[u * 16 + v], h0[v], s);
      th[(size_t)n * 112 + u] = PW_H0 * (h0[u] * s + INV3 * td[u]);
    }
    for (int j = 0; j < 32; j++) {
      float sb = 0.f, sc = 0.f;
#pragma unroll
      for (int u = 0; u < 16; u++) {
        sb = fmaf(Wb[u * 32 + j], h0[u], sb);
        sc = fmaf(Wc[j * 16 + u], h0[u], sc);
      }
      float f = PW_H1 * INV3 * (sb + sc);
      th[(size_t)n * 112 + 16 + j * 3 + 0] = f * sh1[t][j * 3 + 0];
      th[(size_t)n * 112 + 16 + j * 3 + 1] = f * sh1[t][j * 3 + 1];
      th[(size_t)n * 112 + 16 + j * 3 + 2] = f * sh1[t][j * 3 + 2];
    }
  }
}

// ===================== K4: final gather + linear ======================
__global__ __launch_bounds__(256) void out_kernel(
    const float* __restrict__ xg, const float* __restrict__ th,
    const float* __restrict__ WL0, const float* __restrict__ WL1,
    const int* __restrict__ z, const int* __restrict__ canon,
    float* __restrict__ out) {
  __shared__ float sy[256][112];
  int t = threadIdx.x;
  int n = blockIdx.x * 256 + t;
  if (n >= N_NODES) return;
  const float* y = (z[n] > 1) ? (th + (size_t)canon[n] * 112) : (xg + (size_t)n * 112);
  for (int j = 0; j < 112; j++) sy[t][j] = y[j];
#pragma unroll
  for (int w = 0; w < 16; w++) {
    float s = 0.f;
#pragma unroll
    for (int u = 0; u < 16; u++) s = fmaf(sy[t][u], WL0[u * 16 + w], s);
    out[(size_t)n * 112 + w] = 0.25f * s;
  }
  for (int w = 0; w < 32; w++) {
    float s0 = 0.f, s1 = 0.f, s2 = 0.f;
    for (int u = 0; u < 32; u++) {
      float wl = WL1[u * 32 + w];
      s0 = fmaf(sy[t][16 + u * 3 + 0], wl, s0);
      s1 = fmaf(sy[t][16 + u * 3 + 1], wl, s1);
      s2 = fmaf(sy[t][16 + u * 3 + 2], wl, s2);
    }
    out[(size_t)n * 112 + 16 + w * 3 + 0] = RS_L1 * s0;
    out[(size_t)n * 112 + 16 + w * 3 + 1] = RS_L1 * s1;
    out[(size_t)n * 112 + 16 + w * 3 + 2] = RS_L1 * s2;
  }
}

// ============================ launch ============================
extern "C" void kernel_launch(void* const* d_in, const int* in_sizes, int n_in,
                              void* d_out, int out_size, void* d_ws, size_t ws_size,
                              hipStream_t stream) {
  const float* x     = (const float*)d_in[0];
  const float* ea    = (const float*)d_in[1];
  const float* W1    = (const float*)d_in[2];
  const float* W2    = (const float*)d_in[3];
  const float* Wg000 = (const float*)d_in[4];
  const float* Wg110 = (const float*)d_in[5];
  const float* Wg001 = (const float*)d_in[6];
  const float* Wg111 = (const float*)d_in[7];
  const float* Wg012 = (const float*)d_in[8];
  const float* Wg102 = (const float*)d_in[9];
  const float* Wa    = (const float*)d_in[10];
  const float* Wb    = (const float*)d_in[11];
  const float* Wc    = (const float*)d_in[12];
  const float* Wd    = (const float*)d_in[13];
  const float* WL0   = (const float*)d_in[14];
  const float* WL1   = (const float*)d_in[15];
  const int*   eidx  = (const int*)d_in[16];
  const int*   z     = (const int*)d_in[17];
  const int*   canon = (const int*)d_in[18];

  float* ws   = (float*)d_ws;
  float* nm   = ws;                       // 896000
  float* xg   = ws + 896000;              // 896000
  float* sums = ws + 2 * 896000;          // 896000
  float* cnt  = ws + 3 * 896000;          // 8000
  float* th   = ws + 3 * 896000 + 8000;   // 896000

  init_kernel<<<3500, 256, 0, stream>>>(nm, sums, cnt);
  edge_kernel<<<N_EDGES / 16, 256, 0, stream>>>(x, ea, W1, W2, eidx, nm);
  node_gate_kernel<<<N_NODES / 16, 256, 0, stream>>>(x, nm, Wg000, Wg110, Wg001,
                                                     Wg111, Wg012, Wg102, z, canon,
                                                     xg, sums, cnt);
  th_kernel<<<(N_NODES + 255) / 256, 256, 0, stream>>>(sums, cnt, Wa, Wb, Wc, Wd, th);
  out_kernel<<<(N_NODES + 255) / 256, 256, 0, stream>>>(xg, th, WL0, WL1, z, canon,
                                                        (float*)d_out);
}